// Transformer_51445118271695
// MI455X (gfx1250) — compile-verified
//
#include <hip/hip_runtime.h>

typedef float v2f __attribute__((ext_vector_type(2)));
typedef float v8f __attribute__((ext_vector_type(8)));

#define S_LEN   2048
#define DH      64
#define MASK_W  (S_LEN / 32)   // 64 words per mask row
#define NW      4              // waves per attention block (split-K factor)

#if defined(__has_builtin)
#if __has_builtin(__builtin_amdgcn_global_load_async_to_lds_b128)
#define HAVE_ASYNC_LDS 1
#endif
#endif

#ifdef HAVE_ASYNC_LDS
typedef int v4i_ __attribute__((vector_size(4 * sizeof(int))));   // GCC-vector form, matches builtin
typedef __attribute__((address_space(1))) v4i_* gld_t;            // prints as "__device__ *"
typedef __attribute__((address_space(3))) v4i_* lds_t;            // prints as "__shared__ *"
#define GLB_AS(p) ((gld_t)(unsigned long long)(p))
#define LDS_AS(p) ((lds_t)(unsigned)(unsigned long long)(p))      // low 32b = LDS offset
#if __has_builtin(__builtin_amdgcn_s_wait_asynccnt)
#define WAIT_ASYNC(n) __builtin_amdgcn_s_wait_asynccnt(n)
#else
#define WAIT_ASYNC(n) asm volatile("s_wait_asynccnt %0" ::"i"(n) : "memory")
#endif
#endif

// ---------------------------------------------------------------------------
// Kernel 1: faithful museformer mask scan -> bit-packed [S][S/32] in d_ws.
// ---------------------------------------------------------------------------
__device__ __forceinline__ unsigned range_bits(int a, int b, int lo_col, int hi_col) {
  int aa = a > lo_col ? a : lo_col;
  int bb = b < hi_col ? b : hi_col;
  if (aa >= bb) return 0u;
  int n = bb - aa;
  unsigned m = (n >= 32) ? 0xFFFFFFFFu : ((1u << n) - 1u);
  return m << (aa - lo_col);
}

__global__ void muse_mask_kernel(const int* __restrict__ sp, int nb,
                                 unsigned* __restrict__ mbits) {
  __shared__ int ssp[256];
  const int w = threadIdx.x;                  // word index 0..63
  for (int i = threadIdx.x; i < nb; i += blockDim.x) ssp[i] = sp[i];
  __syncthreads();

  const int lo_col = w * 32;
  const int hi_col = lo_col + 32;
  unsigned prev = 0u;
  int bar_count = 0;
  bool first = false;

  for (int i = 0; i < S_LEN; ++i) {
    int bc = bar_count < (nb - 1) ? bar_count : (nb - 1);
    bool is_summary = (bar_count < nb) && (ssp[bc] == i);
    unsigned temp;
    if (is_summary) {
      int lo = (bar_count == 0) ? 1 : ssp[bar_count - 1] + 1;
      temp = range_bits(lo, i, lo_col, hi_col);
    } else if (first) {
      unsigned t = 0u;
#pragma unroll
      for (int si = 0; si < 6; ++si) {
        int s = 1 << si;                       // STRUCT_BARS = 1,2,4,8,16,32
        if (bar_count >= s) {
          int pcol = ssp[bar_count - s];
          if (pcol >= lo_col && pcol < hi_col) t |= 1u << (pcol - lo_col);
        }
        if (bar_count > s) {
          int start = ssp[bar_count - s - 1] + 1;
          int end   = ssp[bar_count - s] - 1;
          t |= range_bits(start, end, lo_col, hi_col);
        }
      }
      temp = t;
    } else {
      temp = prev;
      if (i >= lo_col && i < hi_col) temp |= 1u << (i - lo_col);
    }
    mbits[i * MASK_W + w] = temp;
    if (!is_summary && !first && i >= 1 && w == (i >> 5)) {
      mbits[(i - 1) * MASK_W + w] = prev | (1u << (i & 31)); // torch view-aliasing
    }
    prev = temp;
    if (is_summary) bar_count++;
    first = is_summary;
  }
}

// ---------------------------------------------------------------------------
// Kernel 2: sparse flash attention with fp32 WMMA (16x16x4), async-LDS
// double-buffered K/V staging, mask-based k-tile skipping, split-K merge.
// ---------------------------------------------------------------------------
__global__ __launch_bounds__(32 * NW) void muse_attn_kernel(
    const float* __restrict__ Q, const float* __restrict__ K,
    const float* __restrict__ V, const unsigned char* __restrict__ pad,
    const unsigned* __restrict__ mbits, float* __restrict__ Out, int H) {
  const int bh   = blockIdx.y;
  const int b    = bh / H;
  const int tid  = threadIdx.x;
  const int lane = tid & 31;
  const int wave = tid >> 5;        // 0..NW-1
  const int li   = lane & 15;       // lane within 16-group
  const int hi   = lane >> 4;       // which half of the wave
  const int q0   = blockIdx.x * 16; // query tile base row

  __shared__ unsigned orw_s[64];
  __shared__ int      act_list[128];
  __shared__ int      act_n_s;
  __shared__ float    kbuf[NW][2][16 * 68];  // double-buffered per-wave K tile
  __shared__ float    vbuf[NW][2][16 * 68];  // double-buffered per-wave V tile
  __shared__ float    pscr[NW][16 * 18];     // per-wave P scratch
  __shared__ float    om_s[NW][16];
  __shared__ float    ol_s[NW][16];
  __shared__ float    obuf[NW][16 * 64];

  const size_t base = (size_t)bh * S_LEN * DH;

  // ---- Phase A: k-tile occupancy for this query tile (deterministic) ----
  for (int w = tid; w < 64; w += blockDim.x) {
    unsigned o = 0u;
#pragma unroll
    for (int r = 0; r < 16; ++r) o |= mbits[(q0 + r) * MASK_W + w];
    orw_s[w] = o;
  }
  __syncthreads();
  if (tid == 0) {
    int n = 0;
    for (int kt = 0; kt < S_LEN / 16; ++kt) {
      unsigned half = (orw_s[kt >> 1] >> ((kt & 1) << 4)) & 0xFFFFu;
      if (half) act_list[n++] = kt;
    }
    act_n_s = n;
  }
  __syncthreads();
  const int nact = act_n_s;

  // ---- Phase B: per-wave flash over its share of active tiles ----
  v2f qa[16];
  {
    const float* qrowp = Q + base + (size_t)(q0 + li) * DH;
#pragma unroll
    for (int c = 0; c < 16; ++c) {
      const float* p = qrowp + c * 4 + 2 * hi;
      qa[c].x = p[0];
      qa[c].y = p[1];
    }
  }
  bool pq[8];
#pragma unroll
  for (int r = 0; r < 8; ++r) pq[r] = pad[b * S_LEN + q0 + r + 8 * hi] != 0;

  v8f acco[4];
#pragma unroll
  for (int d = 0; d < 4; ++d) acco[d] = {};
  float mrun[8], lrun[8];
#pragma unroll
  for (int r = 0; r < 8; ++r) { mrun[r] = -3.0e38f; lrun[r] = 0.0f; }

  const float scale = 0.125f;  // 1/sqrt(64)

#ifdef HAVE_ASYNC_LDS
  // -------- async double-buffered staging --------
  int cur = 0;
  if (wave < nact) {    // prologue: stage first tile into buffer 0
    const int kt = act_list[wave];
    const float* Kg = K + base + (size_t)(kt * 16) * DH;
    const float* Vg = V + base + (size_t)(kt * 16) * DH;
#pragma unroll
    for (int j = 0; j < 8; ++j) {
      int flat = j * 32 + lane, row = flat >> 4, c4 = (flat & 15) << 2;
      __builtin_amdgcn_global_load_async_to_lds_b128(
          GLB_AS(Kg + (size_t)row * DH + c4), LDS_AS(&kbuf[wave][0][row * 68 + c4]), 0, 0);
      __builtin_amdgcn_global_load_async_to_lds_b128(
          GLB_AS(Vg + (size_t)row * DH + c4), LDS_AS(&vbuf[wave][0][row * 68 + c4]), 0, 0);
    }
  }
  for (int ii = wave; ii < nact; ii += NW) {
    const int kt = act_list[ii];
    const int nxt = ii + NW;
    if (nxt < nact) {   // prefetch next tile into the other buffer
      const int kt2 = act_list[nxt];
      const float* Kg = K + base + (size_t)(kt2 * 16) * DH;
      const float* Vg = V + base + (size_t)(kt2 * 16) * DH;
      const int bu = cur ^ 1;
#pragma unroll
      for (int j = 0; j < 8; ++j) {
        int flat = j * 32 + lane, row = flat >> 4, c4 = (flat & 15) << 2;
        __builtin_amdgcn_global_load_async_to_lds_b128(
            GLB_AS(Kg + (size_t)row * DH + c4), LDS_AS(&kbuf[wave][bu][row * 68 + c4]), 0, 0);
        __builtin_amdgcn_global_load_async_to_lds_b128(
            GLB_AS(Vg + (size_t)row * DH + c4), LDS_AS(&vbuf[wave][bu][row * 68 + c4]), 0, 0);
      }
      WAIT_ASYNC(16);   // current tile's 16 loads retired; next's 16 in flight
    } else {
      WAIT_ASYNC(0);
    }
    asm volatile("" ::: "memory");
#else
  // -------- synchronous staging fallback --------
  int cur = 0;
  for (int ii = wave; ii < nact; ii += NW) {
    const int kt = act_list[ii];
#pragma unroll
    for (int j = 0; j < 8; ++j) {
      int flat = j * 32 + lane, row = flat >> 4, c4 = (flat & 15) << 2;
      float4 kq = *(const float4*)(K + base + (size_t)(kt * 16 + row) * DH + c4);
      float4 vq = *(const float4*)(V + base + (size_t)(kt * 16 + row) * DH + c4);
      *(float4*)&kbuf[wave][0][row * 68 + c4] = kq;
      *(float4*)&vbuf[wave][0][row * 68 + c4] = vq;
    }
    asm volatile("s_wait_dscnt 0" ::: "memory");
#endif

    const float* kcur = kbuf[wave][cur];
    const float* vcur = vbuf[wave][cur];

    // ---- scores = Q . K^T (16 fp32 WMMAs, K=64) ----
    v8f acc = {};
#pragma unroll
    for (int c = 0; c < 16; ++c) {
      v2f kb = *(const v2f*)&kcur[li * 68 + c * 4 + 2 * hi];
      acc = __builtin_amdgcn_wmma_f32_16x16x4_f32(false, qa[c], false, kb,
                                                  (short)0, acc, false, false);
    }

    // ---- mask + online softmax ----
    const int kcol = kt * 16 + li;
    const bool pk = pad[b * S_LEN + kcol] != 0;
    float pfac[8];
#pragma unroll
    for (int r = 0; r < 8; ++r) {
      const int qrow = q0 + r + 8 * hi;
      unsigned mw = mbits[qrow * MASK_W + (kt >> 1)];
      bool bit = (((mw >> (((kt & 1) << 4) + li)) & 1u) != 0u) && pq[r] && pk;
      float s = bit ? acc[r] * scale : -1.0e9f;
      float m = s;
      m = fmaxf(m, __shfl_xor(m, 1));
      m = fmaxf(m, __shfl_xor(m, 2));
      m = fmaxf(m, __shfl_xor(m, 4));
      m = fmaxf(m, __shfl_xor(m, 8));
      float mnew = fmaxf(mrun[r], m);
      float p  = __expf(s - mnew);
      float co = __expf(mrun[r] - mnew);
      float ps = p;
      ps += __shfl_xor(ps, 1);
      ps += __shfl_xor(ps, 2);
      ps += __shfl_xor(ps, 4);
      ps += __shfl_xor(ps, 8);
      lrun[r] = lrun[r] * co + ps;
      mrun[r] = mnew;
      pfac[r] = co;
      pscr[wave][(r + 8 * hi) * 18 + li] = p;   // C-layout -> LDS
    }
#pragma unroll
    for (int d = 0; d < 4; ++d)
#pragma unroll
      for (int r = 0; r < 8; ++r) acco[d][r] *= pfac[r];

    asm volatile("s_wait_dscnt 0" ::: "memory");

    v2f pa[4];
#pragma unroll
    for (int c = 0; c < 4; ++c)
      pa[c] = *(const v2f*)&pscr[wave][li * 18 + c * 4 + 2 * hi];

    // ---- O += P . V (16 fp32 WMMAs) ----
#pragma unroll
    for (int d = 0; d < 4; ++d) {
#pragma unroll
      for (int c = 0; c < 4; ++c) {
        v2f vb;
        vb.x = vcur[(c * 4 + 2 * hi) * 68 + d * 16 + li];
        vb.y = vcur[(c * 4 + 1 + 2 * hi) * 68 + d * 16 + li];
        acco[d] = __builtin_amdgcn_wmma_f32_16x16x4_f32(false, pa[c], false, vb,
                                                        (short)0, acco[d], false, false);
      }
    }
#ifdef HAVE_ASYNC_LDS
    cur ^= 1;
#endif
  }

  // publish per-wave partials
  if (li == 0) {
#pragma unroll
    for (int r = 0; r < 8; ++r) {
      om_s[wave][r + 8 * hi] = mrun[r];
      ol_s[wave][r + 8 * hi] = lrun[r];
    }
  }
#pragma unroll
  for (int d = 0; d < 4; ++d)
#pragma unroll
    for (int r = 0; r < 8; ++r)
      obuf[wave][(r + 8 * hi) * 64 + d * 16 + li] = acco[d][r];
  __syncthreads();

  // ---- Phase C: deterministic flash merge across the NW waves ----
  for (int e = tid; e < 16 * 64; e += blockDim.x) {
    int row = e >> 6;
    int dim = e & 63;
    float M = -3.0e38f;
#pragma unroll
    for (int w = 0; w < NW; ++w) M = fmaxf(M, om_s[w][row]);
    float L = 0.0f, o = 0.0f;
#pragma unroll
    for (int w = 0; w < NW; ++w) {
      float f = __expf(om_s[w][row] - M);
      L += ol_s[w][row] * f;
      o += obuf[w][row * 64 + dim] * f;
    }
    Out[base + (size_t)(q0 + row) * DH + dim] = o / L;
  }
}

// ---------------------------------------------------------------------------
extern "C" void kernel_launch(void* const* d_in, const int* in_sizes, int n_in,
                              void* d_out, int out_size, void* d_ws, size_t ws_size,
                              hipStream_t stream) {
  const float*         q      = (const float*)d_in[0];
  const float*         k      = (const float*)d_in[1];
  const float*         v      = (const float*)d_in[2];
  const unsigned char* pad    = (const unsigned char*)d_in[3];
  const int*           barpos = (const int*)d_in[4];
  const int nb = in_sizes[4];

  const int BH = in_sizes[0] / (S_LEN * DH);      // B*H
  const int B  = in_sizes[3] / S_LEN;             // batch
  const int H  = BH / B;

  unsigned* mbits = (unsigned*)d_ws;              // 2048*64 u32 = 512 KB
  float* out = (float*)d_out;

  muse_mask_kernel<<<1, MASK_W, 0, stream>>>(barpos, nb, mbits);

  dim3 grid(S_LEN / 16, BH);                      // 128 x BH blocks
  muse_attn_kernel<<<grid, 32 * NW, 0, stream>>>(q, k, v, pad, mbits, out, H);
}